// KeyValueCaching_12163347382340
// MI455X (gfx1250) — compile-verified
//
#include <hip/hip_runtime.h>

// KV-cache concat: out_k = cat(cached_k, new_k, axis=1), out_v likewise.
// B=4, S=4096, NEW=16, D=4096, fp32. Pure bandwidth problem (~1 GiB moved,
// ~44 us floor at 23.3 TB/s). One 256-thread block per output row; the row
// (16 KB) streams HBM -> LDS -> HBM via CDNA5 async global<->LDS b128 ops
// (ASYNCcnt), never touching VGPRs. 3D grid avoids integer-division SALU;
// the shared 24-bit INST_OFFSET (applied to both global and LDS addresses
// per the ISA async pseudocode) lets all 4 chunks reuse one address pair.

#define KB_B    4
#define KB_S    4096
#define KB_NEW  16
#define KB_D    4096
#define KB_ROWS (KB_S + KB_NEW)       /* 4112 rows per batch per tensor */
#define KB_THREADS 256
#define KB_CHUNK_BYTES (KB_THREADS * 16)  /* 4096 B per chunk per block */

typedef int __attribute__((ext_vector_type(4))) v4i;
typedef __attribute__((address_space(1))) v4i* gptr_v4i;  // global
typedef __attribute__((address_space(3))) v4i* lptr_v4i;  // LDS

#if __has_builtin(__builtin_amdgcn_global_load_async_to_lds_b128) && \
    __has_builtin(__builtin_amdgcn_global_store_async_from_lds_b128)
#define KB_USE_ASYNC 1
#else
#define KB_USE_ASYNC 0
#endif

__device__ __forceinline__ gptr_v4i kb_as_global(const void* p) {
    return (gptr_v4i)(v4i*)p;   // drop const, retype, addrspacecast 0->1
}
__device__ __forceinline__ lptr_v4i kb_as_lds(void* p) {
    return (lptr_v4i)(v4i*)p;   // addrspacecast 0->3
}

__global__ __launch_bounds__(KB_THREADS) void kv_concat_kernel(
    const float* __restrict__ cached_k,
    const float* __restrict__ cached_v,
    const float* __restrict__ new_k,
    const float* __restrict__ new_v,
    float* __restrict__ out)
{
    const int row = blockIdx.x;   // 0 .. 4111
    const int b   = blockIdx.y;   // 0 .. 3
    const int t   = blockIdx.z;   // 0 = K, 1 = V

    const float* src;
    if (row < KB_S) {
        const float* base = (t == 0) ? cached_k : cached_v;
        src = base + ((size_t)b * KB_S + (size_t)row) * KB_D;
    } else {
        const float* base = (t == 0) ? new_k : new_v;
        src = base + ((size_t)b * KB_NEW + (size_t)(row - KB_S)) * KB_D;
    }
    float* dst = out + ((size_t)t * KB_B * KB_ROWS + (size_t)b * KB_ROWS
                        + (size_t)row) * KB_D;

    const int tid = threadIdx.x;
    const v4i* src4 = (const v4i*)src + tid;
    v4i*       dst4 = (v4i*)dst + tid;

#if KB_USE_ASYNC
    // 16 KB row staged through LDS; 4 b128 chunks per thread addressed via
    // the instruction immediate offset (applies to global AND LDS sides).
    __shared__ alignas(16) v4i lds_buf[KB_THREADS * 4];

    const gptr_v4i gsrc = kb_as_global(src4);
    const lptr_v4i lbuf = kb_as_lds(&lds_buf[tid]);

    __builtin_amdgcn_global_load_async_to_lds_b128(gsrc, lbuf, 0 * KB_CHUNK_BYTES, 0);
    __builtin_amdgcn_global_load_async_to_lds_b128(gsrc, lbuf, 1 * KB_CHUNK_BYTES, 0);
    __builtin_amdgcn_global_load_async_to_lds_b128(gsrc, lbuf, 2 * KB_CHUNK_BYTES, 0);
    __builtin_amdgcn_global_load_async_to_lds_b128(gsrc, lbuf, 3 * KB_CHUNK_BYTES, 0);

    // Drain async loads into LDS before streaming LDS back out.
#if __has_builtin(__builtin_amdgcn_s_wait_asynccnt)
    __builtin_amdgcn_s_wait_asynccnt(0);
#else
    asm volatile("s_wait_asynccnt 0" ::: "memory");
#endif

    const gptr_v4i gdst = kb_as_global(dst4);

    __builtin_amdgcn_global_store_async_from_lds_b128(gdst, lbuf, 0 * KB_CHUNK_BYTES, 0);
    __builtin_amdgcn_global_store_async_from_lds_b128(gdst, lbuf, 1 * KB_CHUNK_BYTES, 0);
    __builtin_amdgcn_global_store_async_from_lds_b128(gdst, lbuf, 2 * KB_CHUNK_BYTES, 0);
    __builtin_amdgcn_global_store_async_from_lds_b128(gdst, lbuf, 3 * KB_CHUNK_BYTES, 0);
    // s_endpgm's implicit wait-idle drains the outstanding async stores.
#else
    // Fallback: plain fully-coalesced float4 register copy (b128 ld/st).
#pragma unroll
    for (int c = 0; c < 4; ++c) {
        dst4[c * KB_THREADS] = src4[c * KB_THREADS];
    }
#endif
}

extern "C" void kernel_launch(void* const* d_in, const int* in_sizes, int n_in,
                              void* d_out, int out_size, void* d_ws, size_t ws_size,
                              hipStream_t stream) {
    (void)in_sizes; (void)n_in; (void)out_size; (void)d_ws; (void)ws_size;
    const float* cached_k = (const float*)d_in[0];
    const float* cached_v = (const float*)d_in[1];
    const float* new_k    = (const float*)d_in[2];
    const float* new_v    = (const float*)d_in[3];
    float* out = (float*)d_out;

    kv_concat_kernel<<<dim3(KB_ROWS, KB_B, 2), dim3(KB_THREADS), 0, stream>>>(
        cached_k, cached_v, new_k, new_v, out);
}